// LearnableLabelSmoothing_45973329936866
// MI455X (gfx1250) — compile-verified
//
#include <hip/hip_runtime.h>

#define C_CLASSES 1000
#define SPB 8           // samples (output rows) per block

// ---------------------------------------------------------------------------
// Kernel 1: per-class prep.
// One block per class t: rowsum_t = sum_{c != t} relu(omega_d[t,c]) + 1e-8
//   scale[t] = clamp(omega_s[t], 0, 0.5) / rowsum_t
//   diag[t]  = 1 - clamp(omega_s[t], 0, 0.5)
// Tiny kernel (1000 blocks), negligible vs the 256 MB store phase.
// ---------------------------------------------------------------------------
__global__ void __launch_bounds__(256) lls_prep_kernel(
    const float* __restrict__ omega_s,
    const float* __restrict__ omega_d,
    float* __restrict__ scale,
    float* __restrict__ diag)
{
    __shared__ float red[256];
    const int t   = blockIdx.x;
    const int tid = threadIdx.x;
    const float* row = omega_d + (size_t)t * C_CLASSES;

    float acc = 0.0f;
    for (int c = tid; c < C_CLASSES; c += 256) {
        float v = fmaxf(row[c], 0.0f);
        if (c == t) v = 0.0f;          // diagonal masked
        acc += v;
    }
    red[tid] = acc;
    __syncthreads();
    for (int s = 128; s > 0; s >>= 1) {
        if (tid < s) red[tid] += red[tid + s];
        __syncthreads();
    }
    if (tid == 0) {
        float sum = red[0] + 1e-8f;
        float w   = fminf(fmaxf(omega_s[t], 0.0f), 0.5f);
        scale[t]  = w / sum;
        diag[t]   = 1.0f - w;
    }
}

// ---------------------------------------------------------------------------
// Kernel 2: per-sample row materialization (store-bandwidth bound).
// Each block handles SPB samples. For each sample, lane i stages its own
// float4 of omega_d row t into LDS with the CDNA5 async global->LDS mover
// (8 transfers in flight per lane for memory-level parallelism). Because a
// lane reads back exactly the LDS bytes it wrote, a wave-local
// s_wait_asynccnt suffices -- no workgroup barrier needed. Then each lane
// emits relu*scale with the diagonal lane overridden, as a 128-bit store.
// ---------------------------------------------------------------------------
__global__ void __launch_bounds__(256) lls_gather_kernel(
    const int*   __restrict__ targets,
    const float* __restrict__ omega_d,
    const float* __restrict__ scale,
    const float* __restrict__ diag,
    float*       __restrict__ out,
    int batch)
{
    __shared__ __align__(16) float buf[SPB][C_CLASSES];   // 32000 B staging

    const int b0 = blockIdx.x * SPB;
    const int i  = threadIdx.x;          // float4 index 0..249 (250*4 = 1000)

    int t[SPB];
    #pragma unroll
    for (int j = 0; j < SPB; ++j) {
        const int b = b0 + j;
        t[j] = (b < batch) ? targets[b] : 0;   // uniform -> scalar loads
    }

    if (i < 250) {
        #pragma unroll
        for (int j = 0; j < SPB; ++j) {
            unsigned     lds_off = (unsigned)(size_t)(&buf[j][i * 4]);
            const float* gsrc    = omega_d + (size_t)t[j] * C_CLASSES + i * 4;
            // CDNA5 async path: L2 -> LDS, no VGPR round trip (ASYNCcnt).
            asm volatile("global_load_async_to_lds_b128 %0, %1, off"
                         :
                         : "v"(lds_off), "v"(gsrc)
                         : "memory");
        }
    }
    // Wave-local wait: each lane reads back only the bytes it wrote.
    asm volatile("s_wait_asynccnt 0" ::: "memory");

    if (i < 250) {
        const int base = i * 4;
        #pragma unroll
        for (int j = 0; j < SPB; ++j) {
            const int b = b0 + j;
            if (b >= batch) break;

            const float4 v = *(const float4*)(&buf[j][base]);
            const float  s = scale[t[j]];
            float4 o;
            o.x = fmaxf(v.x, 0.0f) * s;
            o.y = fmaxf(v.y, 0.0f) * s;
            o.z = fmaxf(v.z, 0.0f) * s;
            o.w = fmaxf(v.w, 0.0f) * s;

            if ((unsigned)(t[j] - base) < 4u) {  // this float4 holds column t
                const float d = diag[t[j]];
                if      (t[j] == base    ) o.x = d;
                else if (t[j] == base + 1) o.y = d;
                else if (t[j] == base + 2) o.z = d;
                else                       o.w = d;
            }
            *(float4*)(out + (size_t)b * C_CLASSES + base) = o;  // b128 store
        }
    }
}

extern "C" void kernel_launch(void* const* d_in, const int* in_sizes, int n_in,
                              void* d_out, int out_size, void* d_ws, size_t ws_size,
                              hipStream_t stream) {
    const int*   targets = (const int*)  d_in[0];
    const float* omega_s = (const float*)d_in[1];
    const float* omega_d = (const float*)d_in[2];
    float*       out     = (float*)      d_out;

    const int batch = in_sizes[0];

    // Workspace layout: scale[C], diag[C]  (8 KB total)
    float* scale = (float*)d_ws;
    float* diag  = scale + C_CLASSES;

    lls_prep_kernel<<<C_CLASSES, 256, 0, stream>>>(omega_s, omega_d, scale, diag);

    const int nblocks = (batch + SPB - 1) / SPB;
    lls_gather_kernel<<<nblocks, 256, 0, stream>>>(targets, omega_d, scale, diag,
                                                   out, batch);
}